// FCR_41704132444314
// MI455X (gfx1250) — compile-verified
//
#include <hip/hip_runtime.h>
#include <math.h>

// Problem constants (from reference)
#define B_SZ     64
#define C_SZ     3
#define NLINE    256
#define HW       (NLINE * NLINE)          // 65536
#define NIMG     (B_SZ * C_SZ)            // 192 images per tensor
#define EPS_NORM 1e-8f
#define EPS_DEN  1e-7f
#define SCALE    0.01f
#define TWO_PI   6.28318530717958647692f

typedef __attribute__((ext_vector_type(2))) float v2f;
typedef __attribute__((ext_vector_type(8))) float v8f;
typedef int b128_t __attribute__((vector_size(16)));   // builtin's operand type

// ---- V_WMMA_F32_16X16X4_F32 wrapper.  ISA NEG table: f32 WMMA allows only
// CNeg (no A/B negation), so all bool modifiers must be 0. ----
static __device__ __forceinline__ v8f wmma4(v2f a, v2f b, v8f c) {
  return __builtin_amdgcn_wmma_f32_16x16x4_f32(false, a, false, b, (short)0, c,
                                               false, false);
}

// Async bulk copy global -> LDS (GLOBAL_LOAD_ASYNC_TO_LDS_B128, ASYNCcnt).
static __device__ __forceinline__ void async_cp16(const void* g, void* l) {
  __builtin_amdgcn_global_load_async_to_lds_b128((b128_t*)g, (b128_t*)l, 0, 0);
}

// ---------------------------------------------------------------------------
// Kernel 0: build DFT matrix W[j][k] = exp(-2*pi*i*j*k/256):
//   Wr, Wi, Wn = -Wi   (row-major, for coldft A operands)
//   Wpr, Wpi           (pair-packed [k/2][col]{k,k+1}, for rowdft B operands)
// Also zero the distance accumulators.
// ---------------------------------------------------------------------------
__global__ void dft_init(float* __restrict__ Wr, float* __restrict__ Wi,
                         float* __restrict__ Wn, float* __restrict__ Wpr,
                         float* __restrict__ Wpi, float* __restrict__ acc) {
  const int j = blockIdx.x;       // 0..255 (row / k index)
  const int k = threadIdx.x;      // 0..255 (col)
  const int t = (j * k) & (NLINE - 1);
  const float ang = -TWO_PI * (float)t * (1.0f / (float)NLINE);
  float s, c;
  sincosf(ang, &s, &c);
  Wr[j * NLINE + k] = c;
  Wi[j * NLINE + k] = s;
  Wn[j * NLINE + k] = -s;
  const int pk = ((j >> 1) * NLINE + k) * 2 + (j & 1);
  Wpr[pk] = c;
  Wpi[pk] = s;
  if (blockIdx.x == 0 && threadIdx.x < 3 * B_SZ) acc[threadIdx.x] = 0.0f;
}

// ---------------------------------------------------------------------------
// Kernel 1: row DFT.  Y = X * W  (real X, complex W).  One WG per 16-row slab;
// 16 waves, wave w owns output columns [16w, 16w+16).  WMMA f32 16x16x4.
// X slab staged via async global->LDS DMA.
// ---------------------------------------------------------------------------
__global__ __launch_bounds__(512) void rowdft(const float* __restrict__ X,
                                              float2* __restrict__ F,
                                              const float2* __restrict__ Wpr,
                                              const float2* __restrict__ Wpi) {
  const int img = blockIdx.x >> 4;
  const int rb  = blockIdx.x & 15;
  const float* x = X + (size_t)img * HW + (size_t)rb * 16 * NLINE;
  float2*      y = F + (size_t)img * HW + (size_t)rb * 16 * NLINE;

  __shared__ float xs[16 * NLINE];  // 16 KB slab of X
  // 1024 x b128 async transfers, 2 per thread
  for (int i = threadIdx.x; i < 1024; i += 512) {
    async_cp16(x + i * 4, xs + i * 4);
  }
  __builtin_amdgcn_s_wait_asynccnt(0);
  __syncthreads();

  const int wave = threadIdx.x >> 5;
  const int lane = threadIdx.x & 31;
  const int half = lane >> 4;       // ISA 16x4 A layout: K pairs split by half
  const int l16  = lane & 15;
  const int col0 = wave << 4;

  v8f cr = {};
  v8f ci = {};
  const float* xrow = &xs[l16 * NLINE];

  for (int k = 0; k < NLINE; k += 4) {
    const int kk = k + (half << 1);
    const int pidx = (kk >> 1) * NLINE + col0 + l16;  // pair-packed index
    v2f a;  a.x  = xrow[kk];
            a.y  = xrow[kk + 1];
    const float2 tr = Wpr[pidx];    // {W[kk][col], W[kk+1][col]} one b64
    const float2 ti = Wpi[pidx];
    v2f br; br.x = tr.x; br.y = tr.y;
    v2f bi; bi.x = ti.x; bi.y = ti.y;
    cr = wmma4(a, br, cr);
    ci = wmma4(a, bi, ci);
  }

#pragma unroll
  for (int v = 0; v < 8; ++v) {
    const int m = (half << 3) + v;  // C/D layout: row = vgpr + 8*half
    y[(size_t)m * NLINE + col0 + l16] = make_float2(cr[v], ci[v]);
  }
}

// ---------------------------------------------------------------------------
// Kernel 2: column DFT, IN PLACE.  F = W * Y (complex*complex).  One WG per
// 16-column slab (columns are independent under the column transform); slab
// staged interleaved (float2) via async DMA.
// Fr = Wr*Yr + Wn*Yi (Wn = -Wi),  Fi = Wr*Yi + Wi*Yr.
// ---------------------------------------------------------------------------
__global__ __launch_bounds__(512) void coldft(float2* __restrict__ F,
                                              const float* __restrict__ Wr,
                                              const float* __restrict__ Wi,
                                              const float* __restrict__ Wn) {
  const int img = blockIdx.x >> 4;
  const int cb  = blockIdx.x & 15;
  float2* y = F + (size_t)img * HW + (size_t)(cb << 4);

  __shared__ float2 ys[NLINE * 16];  // 32 KB interleaved complex slab
  // 2048 x b128 async transfers (2 float2 each), 4 per thread
  for (int i = threadIdx.x; i < 2048; i += 512) {
    const int r = i >> 3, seg = i & 7;
    async_cp16(y + (size_t)r * NLINE + seg * 2, ys + r * 16 + seg * 2);
  }
  __builtin_amdgcn_s_wait_asynccnt(0);
  __syncthreads();

  const int wave = threadIdx.x >> 5;
  const int lane = threadIdx.x & 31;
  const int half = lane >> 4;
  const int l16  = lane & 15;
  const int row0 = wave << 4;

  v8f fr = {};
  v8f fi = {};
  const float2* wr2 = (const float2*)&Wr[(size_t)(row0 + l16) * NLINE];
  const float2* wi2 = (const float2*)&Wi[(size_t)(row0 + l16) * NLINE];
  const float2* wn2 = (const float2*)&Wn[(size_t)(row0 + l16) * NLINE];

  for (int k = 0; k < NLINE; k += 4) {
    const int kk = k + (half << 1);
    const float2 tr = wr2[kk >> 1];  // {W[row][kk], W[row][kk+1]} one b64
    const float2 ti = wi2[kk >> 1];
    const float2 tn = wn2[kk >> 1];
    v2f ar; ar.x = tr.x; ar.y = tr.y;
    v2f ai; ai.x = ti.x; ai.y = ti.y;
    v2f an; an.x = tn.x; an.y = tn.y;
    const float2 y0 = ys[kk * 16 + l16];        // one ds_load_b64: {Yr,Yi}
    const float2 y1 = ys[(kk + 1) * 16 + l16];
    v2f br; br.x = y0.x; br.y = y1.x;
    v2f bi; bi.x = y0.y; bi.y = y1.y;
    fr = wmma4(ar, br, fr);
    fr = wmma4(an, bi, fr);        // Fr = Wr*Yr - Wi*Yi
    fi = wmma4(ar, bi, fi);
    fi = wmma4(ai, br, fi);        // Fi = Wr*Yi + Wi*Yr
  }

#pragma unroll
  for (int v = 0; v < 8; ++v) {
    const int m = row0 + (half << 3) + v;
    y[(size_t)m * NLINE + l16] = make_float2(fr[v], fi[v]);
  }
}

// ---------------------------------------------------------------------------
// Kernel 3: fused normalize + complex-L1 triplet distances.
// acc[b*3+0] = sum |af-pf| ; acc[b*3+1] = sum |af-nf[j0]| ; +2 = j1.
// ---------------------------------------------------------------------------
__global__ __launch_bounds__(256) void reduce_dist(const float2* __restrict__ Fa,
                                                   const float2* __restrict__ Fp,
                                                   const float2* __restrict__ Fn,
                                                   const int* __restrict__ neg_idx,
                                                   float* __restrict__ acc) {
  const int b     = blockIdx.x >> 6;
  const int chunk = blockIdx.x & 63;
  const int j0 = neg_idx[2 * b + 0];
  const int j1 = neg_idx[2 * b + 1];

  float s_ap = 0.0f, s_n0 = 0.0f, s_n1 = 0.0f;

  for (int p = 0; p < 4; ++p) {
    const int hw = (chunk << 10) + (p << 8) + threadIdx.x;
    float2 fa[3], fp2[3], f0[3], f1[3];
    float na = 0.f, npp = 0.f, n0 = 0.f, n1 = 0.f;
#pragma unroll
    for (int c = 0; c < 3; ++c) {
      fa[c]  = Fa[((size_t)(b * 3 + c)) * HW + hw];
      fp2[c] = Fp[((size_t)(b * 3 + c)) * HW + hw];
      f0[c]  = Fn[((size_t)(j0 * 3 + c)) * HW + hw];
      f1[c]  = Fn[((size_t)(j1 * 3 + c)) * HW + hw];
      na  += fa[c].x * fa[c].x + fa[c].y * fa[c].y;
      npp += fp2[c].x * fp2[c].x + fp2[c].y * fp2[c].y;
      n0  += f0[c].x * f0[c].x + f0[c].y * f0[c].y;
      n1  += f1[c].x * f1[c].x + f1[c].y * f1[c].y;
    }
    const float sa  = SCALE / (sqrtf(na) + EPS_NORM);
    const float sp  = SCALE / (sqrtf(npp) + EPS_NORM);
    const float sn0 = SCALE / (sqrtf(n0) + EPS_NORM);
    const float sn1 = SCALE / (sqrtf(n1) + EPS_NORM);
#pragma unroll
    for (int c = 0; c < 3; ++c) {
      float dx = fa[c].x * sa - fp2[c].x * sp;
      float dy = fa[c].y * sa - fp2[c].y * sp;
      s_ap += sqrtf(dx * dx + dy * dy);
      dx = fa[c].x * sa - f0[c].x * sn0;
      dy = fa[c].y * sa - f0[c].y * sn0;
      s_n0 += sqrtf(dx * dx + dy * dy);
      dx = fa[c].x * sa - f1[c].x * sn1;
      dy = fa[c].y * sa - f1[c].y * sn1;
      s_n1 += sqrtf(dx * dx + dy * dy);
    }
  }

  __shared__ float r0[256], r1[256], r2[256];
  r0[threadIdx.x] = s_ap;
  r1[threadIdx.x] = s_n0;
  r2[threadIdx.x] = s_n1;
  __syncthreads();
  for (int s = 128; s > 0; s >>= 1) {
    if (threadIdx.x < s) {
      r0[threadIdx.x] += r0[threadIdx.x + s];
      r1[threadIdx.x] += r1[threadIdx.x + s];
      r2[threadIdx.x] += r2[threadIdx.x + s];
    }
    __syncthreads();
  }
  if (threadIdx.x == 0) {
    atomicAdd(&acc[b * 3 + 0], r0[0]);
    atomicAdd(&acc[b * 3 + 1], r1[0]);
    atomicAdd(&acc[b * 3 + 2], r2[0]);
  }
}

// ---------------------------------------------------------------------------
// Kernel 4: final contrastive sum -> scalar
// ---------------------------------------------------------------------------
__global__ void finalize(const float* __restrict__ acc, float* __restrict__ out) {
  __shared__ float red[64];
  const int b = threadIdx.x;
  const float inv = 1.0f / ((float)C_SZ * (float)HW);
  const float dap = acc[b * 3 + 0] * inv;
  const float v = dap / (acc[b * 3 + 1] * inv + EPS_DEN) +
                  dap / (acc[b * 3 + 2] * inv + EPS_DEN);
  red[b] = v;
  __syncthreads();
  for (int s = 32; s > 0; s >>= 1) {
    if (b < s) red[b] += red[b + s];
    __syncthreads();
  }
  if (b == 0) out[0] = red[0] * (1.0f / (2.0f * (float)B_SZ));
}

// ---------------------------------------------------------------------------
extern "C" void kernel_launch(void* const* d_in, const int* in_sizes, int n_in,
                              void* d_out, int out_size, void* d_ws, size_t ws_size,
                              hipStream_t stream) {
  (void)in_sizes; (void)n_in; (void)out_size; (void)ws_size;
  const float* a = (const float*)d_in[0];
  const float* p = (const float*)d_in[1];
  const float* n = (const float*)d_in[2];
  const int* neg_idx = (const int*)d_in[3];
  float* out = (float*)d_out;

  char* ws = (char*)d_ws;
  float*  Wr  = (float*)(ws);                       // 256 KB row-major cos
  float*  Wi  = (float*)(ws + 262144);              // 256 KB row-major -sin
  float*  Wn  = (float*)(ws + 524288);              // 256 KB row-major +sin (-Wi)
  float2* Wpr = (float2*)(ws + 786432);             // 256 KB pair-packed cos
  float2* Wpi = (float2*)(ws + 1048576);            // 256 KB pair-packed -sin
  float*  acc = (float*)(ws + 1310720);             // 192 floats
  const size_t FBYTES = (size_t)NIMG * HW * sizeof(float2);  // ~96 MB each
  float2* Fa = (float2*)(ws + 1314816);
  float2* Fp = (float2*)(ws + 1314816 + FBYTES);
  float2* Fn = (float2*)(ws + 1314816 + 2 * FBYTES);

  dft_init<<<NLINE, NLINE, 0, stream>>>(Wr, Wi, Wn, (float*)Wpr, (float*)Wpi, acc);

  rowdft<<<NIMG * 16, 512, 0, stream>>>(a, Fa, Wpr, Wpi);
  rowdft<<<NIMG * 16, 512, 0, stream>>>(p, Fp, Wpr, Wpi);
  rowdft<<<NIMG * 16, 512, 0, stream>>>(n, Fn, Wpr, Wpi);

  coldft<<<NIMG * 16, 512, 0, stream>>>(Fa, Wr, Wi, Wn);
  coldft<<<NIMG * 16, 512, 0, stream>>>(Fp, Wr, Wi, Wn);
  coldft<<<NIMG * 16, 512, 0, stream>>>(Fn, Wr, Wi, Wn);

  reduce_dist<<<B_SZ * 64, 256, 0, stream>>>(Fa, Fp, Fn, neg_idx, acc);
  finalize<<<1, 64, 0, stream>>>(acc, out);
}